// CRF_20727512171211
// MI455X (gfx1250) — compile-verified
//
#include <hip/hip_runtime.h>
#include <math.h>

// ---------------------------------------------------------------------------
// CRF forward scan on gfx1250 (MI455X).
//
//   fv_{t+1}[j] = M + log( sum_i exp(fv_t[i]-M) * Eg[j,i] ) + feats[t][j]
//   Eg = exp(transitions)   (4 MB f32, precomputed, LDS-resident in slices)
//
// Design: latency-bound sequential scan -> persistent kernel, 128 WGs,
// each WG holds an 8x1024 f32 slice of Eg in LDS (32 KB, loaded once via
// global_load_async_to_lds_b128 / ASYNCcnt), per-step feats slice prefetched
// one step ahead with global_load_async_to_lds_b32, fv (4 KB) ping-ponged
// through L2 with one atomic grid barrier per step.
//
// Gold-score emission sum uses V_WMMA_F32_16X16X4_F32 as a ones-matmul
// column reduction (bandwidth-bound kernel; matrix pipe does the adds free).
// ---------------------------------------------------------------------------

#define SEQ_LEN 16384
#define TAG     1024
#define NEG_INF (-10000.0f)

#define NWG    128
#define ROWS   (TAG / NWG)      // 8 rows per workgroup
#define BLOCK  256              // 8 wave32s -> one wave per row
#define WAVES  (BLOCK / 32)

// ---- workspace layout (float indices) -------------------------------------
#define EG_F    0                       // exp(transitions): 1024*1024 f32 (4 MB)
#define FV0_F   (TAG * TAG)             // fv ping buffer   : 1024 f32
#define FV1_F   (FV0_F + TAG)           // fv pong buffer   : 1024 f32
#define EMIT_F  (FV0_F + 2 * TAG)       // emit_sum         : 1024 f32
#define FSC_F   (EMIT_F + TAG)          // forward_score    : 1 f32
#define TRS_F   (FSC_F + 1)             // trans_sum        : 1 f32
#define BAR_F   (FSC_F + 8)             // grid barrier     : 2 u32 (count, gen)
// total ~4.21 MB of d_ws

typedef float v2f __attribute__((ext_vector_type(2)));
typedef float v8f __attribute__((ext_vector_type(8)));

// low 32 bits of a generic pointer to __shared__ == LDS byte offset (AS3 addr)
__device__ __forceinline__ unsigned lds_off(const void* p) {
    return (unsigned)(size_t)p;
}

// ---------------------------------------------------------------------------
__global__ void crf_init(float* ws, const int* start_idx) {
    int tid = threadIdx.x;               // <<<1, 1024>>>
    if (tid < TAG) {
        ws[FV0_F + tid] = (tid == start_idx[0]) ? 0.0f : NEG_INF;
        ws[FV1_F + tid] = NEG_INF;
    }
    if (tid == 0) {
        unsigned* bar = (unsigned*)(ws + BAR_F);
        bar[0] = 0u;                     // arrival count
        bar[1] = 0u;                     // generation
        ws[FSC_F] = 0.0f;
        ws[TRS_F] = 0.0f;
    }
}

// Eg = exp(transitions), elementwise over 1M f32 (float4 vectorized)
__global__ void crf_exp(float* ws, const float* trans) {
    int i = blockIdx.x * blockDim.x + threadIdx.x;   // <<<1024, 256>>> -> 262144 float4
    float4 t = ((const float4*)trans)[i];
    float4 e;
    e.x = expf(t.x); e.y = expf(t.y); e.z = expf(t.z); e.w = expf(t.w);
    ((float4*)(ws + EG_F))[i] = e;
}

// ---------------------------------------------------------------------------
__global__ void __launch_bounds__(BLOCK) crf_scan(float* ws,
                                                  const float* feats,
                                                  const float* trans,
                                                  const int* stop_idx) {
    __shared__ float Elds[ROWS][TAG];    // 32 KB slice of exp(transitions)
    __shared__ float aLds[TAG];          // exp(fv - M), shared per step
    __shared__ float red[WAVES];
    __shared__ float fbuf[2][ROWS];      // double-buffered feats slice

    const int tid  = threadIdx.x;
    const int wg   = blockIdx.x;
    const int lane = tid & 31;
    const int w    = tid >> 5;           // wave id == owned row within slice

    float*    fv0 = ws + FV0_F;
    float*    fv1 = ws + FV1_F;
    unsigned* bar = (unsigned*)(ws + BAR_F);
    const float* Eg = ws + EG_F + (size_t)wg * ROWS * TAG;

    // ---- one-time async preload of the Eg slice (b128, 8 iters x 256 thr x 16B)
    const unsigned eBase = lds_off(&Elds[0][0]);
    #pragma unroll
    for (int k = 0; k < (ROWS * TAG * 4) / (BLOCK * 16); ++k) {
        unsigned boff = (unsigned)((k * BLOCK + tid) * 16);
        asm volatile("global_load_async_to_lds_b128 %0, %1, %2"
                     :: "v"(eBase + boff), "v"(boff), "s"(Eg) : "memory");
    }
    // ---- async prefetch of feats slice for t = 0
    const unsigned fBase = lds_off(&fbuf[0][0]);
    if (tid < ROWS) {
        unsigned boff = (unsigned)((wg * ROWS + tid) * 4);
        asm volatile("global_load_async_to_lds_b32 %0, %1, %2"
                     :: "v"(fBase + (unsigned)(tid * 4)), "v"(boff), "s"(feats)
                     : "memory");
    }

    for (int t = 0; t < SEQ_LEN; ++t) {
        float* fvcur = (t & 1) ? fv1 : fv0;
        float* fvnxt = (t & 1) ? fv0 : fv1;

        // -- read full fv (4 contiguous f32 per thread), block max-reduce
        float4 v = ((const float4*)fvcur)[tid];
        float m = fmaxf(fmaxf(v.x, v.y), fmaxf(v.z, v.w));
        #pragma unroll
        for (int off = 16; off; off >>= 1) m = fmaxf(m, __shfl_xor(m, off));
        if (lane == 0) red[w] = m;
        __syncthreads();
        float M = red[0];
        #pragma unroll
        for (int i = 1; i < WAVES; ++i) M = fmaxf(M, red[i]);

        // -- a[i] = exp(fv[i] - M)
        float4 a4;
        a4.x = expf(v.x - M); a4.y = expf(v.y - M);
        a4.z = expf(v.z - M); a4.w = expf(v.w - M);
        ((float4*)aLds)[tid] = a4;

        // async (Eg slice at t==0, feats slice issued last step) must have landed
        asm volatile("s_wait_asynccnt 0" ::: "memory");
        __syncthreads();

        // -- dot: wave w owns row w; lane sums 32 stride-32 terms, shuffle-reduce
        float s = 0.0f;
        const float* Er = &Elds[w][0];
        #pragma unroll 8
        for (int k = 0; k < TAG / 32; ++k) {
            int i = lane + k * 32;
            s = fmaf(aLds[i], Er[i], s);
        }
        #pragma unroll
        for (int off = 16; off; off >>= 1) s += __shfl_xor(s, off);
        if (lane == 0)
            fvnxt[wg * ROWS + w] = M + logf(s) + fbuf[t & 1][w];

        // -- async prefetch next step's feats slice into the other buffer
        if ((t + 1) < SEQ_LEN && tid < ROWS) {
            unsigned boff = (unsigned)(((t + 1) * TAG + wg * ROWS + tid) * 4);
            unsigned loff = fBase + (unsigned)((((t + 1) & 1) * ROWS + tid) * 4);
            asm volatile("global_load_async_to_lds_b32 %0, %1, %2"
                         :: "v"(loff), "v"(boff), "s"(feats) : "memory");
        }

        // -- device-wide barrier (sense-reversing, agent scope)
        __threadfence();
        __syncthreads();
        if (tid == 0) {
            unsigned g = __hip_atomic_load(&bar[1], __ATOMIC_RELAXED,
                                           __HIP_MEMORY_SCOPE_AGENT);
            unsigned arrived = __hip_atomic_fetch_add(&bar[0], 1u, __ATOMIC_ACQ_REL,
                                                      __HIP_MEMORY_SCOPE_AGENT);
            if (arrived == NWG - 1) {
                __hip_atomic_store(&bar[0], 0u, __ATOMIC_RELAXED,
                                   __HIP_MEMORY_SCOPE_AGENT);
                __hip_atomic_fetch_add(&bar[1], 1u, __ATOMIC_RELEASE,
                                       __HIP_MEMORY_SCOPE_AGENT);
            } else {
                while (__hip_atomic_load(&bar[1], __ATOMIC_ACQUIRE,
                                         __HIP_MEMORY_SCOPE_AGENT) == g)
                    __builtin_amdgcn_s_sleep(2);
            }
        }
        __syncthreads();
        __threadfence();
    }

    // ---- epilogue (block 0): forward_score = LSE(fv_final + trans[stop])
    if (wg == 0) {
        const float* fvF = fv0;                       // SEQ_LEN even -> fv0
        const float* Ts  = trans + (size_t)stop_idx[0] * TAG;
        float4 v  = ((const float4*)fvF)[tid];
        float4 tr = ((const float4*)Ts)[tid];
        float x0 = v.x + tr.x, x1 = v.y + tr.y, x2 = v.z + tr.z, x3 = v.w + tr.w;
        float m = fmaxf(fmaxf(x0, x1), fmaxf(x2, x3));
        #pragma unroll
        for (int off = 16; off; off >>= 1) m = fmaxf(m, __shfl_xor(m, off));
        if (lane == 0) red[w] = m;
        __syncthreads();
        float M2 = red[0];
        #pragma unroll
        for (int i = 1; i < WAVES; ++i) M2 = fmaxf(M2, red[i]);
        float se = expf(x0 - M2) + expf(x1 - M2) + expf(x2 - M2) + expf(x3 - M2);
        #pragma unroll
        for (int off = 16; off; off >>= 1) se += __shfl_xor(se, off);
        __syncthreads();
        if (lane == 0) red[w] = se;
        __syncthreads();
        if (tid == 0) {
            float ssum = 0.0f;
            #pragma unroll
            for (int i = 0; i < WAVES; ++i) ssum += red[i];
            ws[FSC_F] = M2 + logf(ssum);
        }
    }
}

// ---------------------------------------------------------------------------
// gold score:
//   blocks 0..7 (64 waves): emit_sum[j] = sum_t feats[tags[t], j]
//     -> WMMA ones-matmul column reduction: D = ones(16x4) x B(4x16) + C
//        gives D[m,n] = sum_k B[k,n]; A=ones makes the A layout irrelevant,
//        and summing over all K makes the B row<->K assignment irrelevant.
//        4 independent accumulators pipeline the matrix unit; the kernel is
//        bandwidth-bound (64 MB stream) so the adds are free.
//   block  8: trans_sum = sum_t T[tags_ext[t+1], tags_ext[t]]
__global__ void __launch_bounds__(256) crf_gold(float* ws, const float* feats,
                                                const float* trans,
                                                const int* tags,
                                                const int* start_idx) {
    const int tid = threadIdx.x;
    if (blockIdx.x < 8) {
        const int w    = tid >> 5;
        const int lane = tid & 31;
        const int jb   = (blockIdx.x * WAVES + w) * 16;  // 16-column block
        const int n    = lane & 15;                      // N index within block
        const int kk   = (lane >> 4) * 2;                // K rows this half feeds
#if __has_builtin(__builtin_amdgcn_wmma_f32_16x16x4_f32)
        v8f c0 = {}, c1 = {}, c2 = {}, c3 = {};
        const v2f ones = {1.0f, 1.0f};
        for (int t = 0; t < SEQ_LEN; t += 16) {
            v2f b0, b1, b2, b3;
            b0.x = feats[(size_t)tags[t +  0 + kk] * TAG + jb + n];
            b0.y = feats[(size_t)tags[t +  1 + kk] * TAG + jb + n];
            b1.x = feats[(size_t)tags[t +  4 + kk] * TAG + jb + n];
            b1.y = feats[(size_t)tags[t +  5 + kk] * TAG + jb + n];
            b2.x = feats[(size_t)tags[t +  8 + kk] * TAG + jb + n];
            b2.y = feats[(size_t)tags[t +  9 + kk] * TAG + jb + n];
            b3.x = feats[(size_t)tags[t + 12 + kk] * TAG + jb + n];
            b3.y = feats[(size_t)tags[t + 13 + kk] * TAG + jb + n];
            c0 = __builtin_amdgcn_wmma_f32_16x16x4_f32(false, ones, false, b0,
                                                       (short)0, c0, false, false);
            c1 = __builtin_amdgcn_wmma_f32_16x16x4_f32(false, ones, false, b1,
                                                       (short)0, c1, false, false);
            c2 = __builtin_amdgcn_wmma_f32_16x16x4_f32(false, ones, false, b2,
                                                       (short)0, c2, false, false);
            c3 = __builtin_amdgcn_wmma_f32_16x16x4_f32(false, ones, false, b3,
                                                       (short)0, c3, false, false);
        }
        // D layout: VGPR0 = row M=0, N=lane (lanes 0..15); all M rows identical.
        if (lane < 16)
            ws[EMIT_F + jb + lane] = c0[0] + c1[0] + c2[0] + c3[0];
#else
        // scalar fallback: 16 active lanes per wave, one column each
        if (lane < 16) {
            int j = jb + lane;
            float acc = 0.0f;
            for (int t = 0; t < SEQ_LEN; ++t)
                acc += feats[(size_t)tags[t] * TAG + j];
            ws[EMIT_F + j] = acc;
        }
#endif
    } else {
        __shared__ float red[256];
        float acc = 0.0f;
        for (int t = (int)threadIdx.x; t < SEQ_LEN; t += 256) {
            int cur  = tags[t];
            int prev = (t == 0) ? start_idx[0] : tags[t - 1];
            acc += trans[(size_t)cur * TAG + prev];
        }
        red[threadIdx.x] = acc;
        __syncthreads();
        for (int s = 128; s; s >>= 1) {
            if ((int)threadIdx.x < s) red[threadIdx.x] += red[threadIdx.x + s];
            __syncthreads();
        }
        if (threadIdx.x == 0) ws[TRS_F] = red[0];
    }
}

// out[j] = forward_score - (trans_sum + emit_sum[j] + T[stop, tags[-1]])
__global__ void crf_final(float* out, const float* ws, const float* trans,
                          const int* tags, const int* stop_idx) {
    int j = blockIdx.x * blockDim.x + threadIdx.x;   // <<<4, 256>>>
    float last = trans[(size_t)stop_idx[0] * TAG + tags[SEQ_LEN - 1]];
    out[j] = ws[FSC_F] - (ws[TRS_F] + ws[EMIT_F + j] + last);
}

// ---------------------------------------------------------------------------
extern "C" void kernel_launch(void* const* d_in, const int* in_sizes, int n_in,
                              void* d_out, int out_size, void* d_ws, size_t ws_size,
                              hipStream_t stream) {
    (void)in_sizes; (void)n_in; (void)out_size; (void)ws_size;
    const float* feats = (const float*)d_in[0];
    const float* trans = (const float*)d_in[1];
    const int*   tags  = (const int*)d_in[2];
    const int*   start = (const int*)d_in[3];
    const int*   stop  = (const int*)d_in[4];
    float* ws  = (float*)d_ws;
    float* out = (float*)d_out;

    crf_init <<<1, 1024, 0, stream>>>(ws, start);
    crf_exp  <<<(TAG * TAG) / (256 * 4), 256, 0, stream>>>(ws, trans);
    crf_scan <<<NWG, BLOCK, 0, stream>>>(ws, feats, trans, stop);
    crf_gold <<<9, 256, 0, stream>>>(ws, feats, trans, tags, start);
    crf_final<<<4, 256, 0, stream>>>(out, ws, trans, tags, stop);
}